// SoftDetectionModule_v4_26688926777531
// MI455X (gfx1250) — compile-verified
//
#include <hip/hip_runtime.h>
#include <hip/hip_bf16.h>

// CDNA5 / gfx1250 fused SoftDetection kernel.
//
// y[2,N] = W[2,64] x[64,N]  via V_WMMA_F32_16X16X4_F32 chains (fp32 in, fp32 acc),
// then exp, 3x3 edge-replicated box sum over both channels in LDS, out = e0/sum.
//
// Memory-bound problem: 268 MB of x reads -> ~12 us floor at 23.3 TB/s.
// Single fused pass; each 256-thread block (8 wave32) handles an 8-row strip
// of one image plus 1 halo row each side (recomputed, +25% reads, no HBM
// round-trip for intermediates, no d_ws dependency).

typedef __attribute__((ext_vector_type(2))) float v2f;
typedef __attribute__((ext_vector_type(8))) float v8f;

#define BDIM 16
#define CDIM 64
#define HDIM 256
#define WDIM 256
#define HW   (HDIM * WDIM)
#define RROWS 8   // interior rows per block

__global__ __launch_bounds__(256, 1) void soft_det_fused(
    const float* __restrict__ x,     // [B, C, H, W]
    const float* __restrict__ Wt,    // [2, C]
    const float* __restrict__ bias,  // [2]
    float* __restrict__ out)         // [B, H, W]
{
    __shared__ float Ssum[(RROWS + 2) * WDIM];  // e0+e1 rows (with halo)
    __shared__ float Se0[RROWS * WDIM];         // e0 interior rows

    const int tid  = threadIdx.x;
    const int lane = tid & 31;
    const int wv   = tid >> 5;
    const int n    = lane & 15;   // pixel-in-group index / A-matrix M row
    const int kh   = lane >> 4;   // which K-half this lane holds (0 or 1)

    const int bimg  = blockIdx.x >> 5;          // 32 strips per image
    const int strip = blockIdx.x & 31;
    const int h0    = strip * RROWS;

    const float b0 = bias[0];
    const float b1 = bias[1];

    // A-fragment source: rows m>=2 of the 16x64 "padded W" are zero.
    // Clamp the pointer so inactive lanes still load in-bounds, then cndmask.
    const bool  wact = (n < 2);
    const float* wrow = Wt + (wact ? n : 0) * CDIM;

    const float* ximg = x + (size_t)bimg * CDIM * HW;
    const int col0 = wv * 32;          // this wave's 32 columns
    const int mycol = tid;             // == col0 + lane

    for (int i = 0; i < RROWS + 2; ++i) {
        int hh = h0 - 1 + i;
        hh = (hh < 0) ? 0 : ((hh > HDIM - 1) ? (HDIM - 1) : hh);  // edge replicate

        // lane base: channel 2*kh, row hh, column col0+n
        const float* xlane = ximg + (size_t)(2 * kh) * HW + hh * WDIM + col0 + n;

        v8f acc_a = {0.f, 0.f, 0.f, 0.f, 0.f, 0.f, 0.f, 0.f};
        v8f acc_b = {0.f, 0.f, 0.f, 0.f, 0.f, 0.f, 0.f, 0.f};

#pragma unroll
        for (int j = 0; j < CDIM / 4; ++j) {
            const size_t off = (size_t)(4 * j) * HW;
            const int k0 = 4 * j + 2 * kh;

            v2f A;
            {
                float wa = wrow[k0];
                float wb = wrow[k0 + 1];
                A.x = wact ? wa : 0.f;
                A.y = wact ? wb : 0.f;
            }
            // B fragments: group A = cols [col0, col0+16), group B = +16
            v2f Ba, Bb;
            Ba.x = xlane[off];
            Ba.y = xlane[off + HW];
            Bb.x = xlane[off + 16];
            Bb.y = xlane[off + HW + 16];

            // D = A x B + C, fp32 accumulate (matches reference numerics)
            acc_a = __builtin_amdgcn_wmma_f32_16x16x4_f32(
                false, A, false, Ba, (short)0, acc_a, false, false);
            acc_b = __builtin_amdgcn_wmma_f32_16x16x4_f32(
                false, A, false, Bb, (short)0, acc_b, false, false);
        }

        // D layout: VGPR0 lanes0-15 = row M=0 (y0), VGPR1 lanes0-15 = row M=1 (y1).
        // Merge the two 16-pixel groups into one 32-lane coalesced vector.
        float y0a = acc_a[0], y1a = acc_a[1];
        float y0b = acc_b[0], y1b = acc_b[1];
        float y0s = __shfl(y0b, n);
        float y1s = __shfl(y1b, n);
        float Y0 = (lane < 16) ? y0a : y0s;
        float Y1 = (lane < 16) ? y1a : y1s;

        float e0 = expf(Y0 + b0);
        float e1 = expf(Y1 + b1);

        Ssum[i * WDIM + mycol] = e0 + e1;
        if (i >= 1 && i <= RROWS)
            Se0[(i - 1) * WDIM + mycol] = e0;
    }

    __syncthreads();

    // 3x3 edge-replicated box sum + divide, one column per thread.
    const int cm = (mycol > 0) ? mycol - 1 : 0;
    const int cp = (mycol < WDIM - 1) ? mycol + 1 : WDIM - 1;
    float* outimg = out + (size_t)bimg * HW;

#pragma unroll
    for (int r = 0; r < RROWS; ++r) {
        float s = 0.f;
#pragma unroll
        for (int dr = 0; dr < 3; ++dr) {
            const float* Srow = &Ssum[(r + dr) * WDIM];
            s += Srow[cm] + Srow[mycol] + Srow[cp];
        }
        outimg[(h0 + r) * WDIM + mycol] = Se0[r * WDIM + mycol] / s;
    }
}

extern "C" void kernel_launch(void* const* d_in, const int* in_sizes, int n_in,
                              void* d_out, int out_size, void* d_ws, size_t ws_size,
                              hipStream_t stream) {
    (void)in_sizes; (void)n_in; (void)out_size; (void)d_ws; (void)ws_size;
    const float* x    = (const float*)d_in[0];
    const float* Wt   = (const float*)d_in[1];
    const float* bias = (const float*)d_in[2];
    float* out = (float*)d_out;

    dim3 grid(BDIM * (HDIM / RROWS));  // 16 images * 32 strips = 512 blocks
    dim3 block(256);                   // 8 wave32 per block
    soft_det_fused<<<grid, block, 0, stream>>>(x, Wt, bias, out);
}